// ClassifierDeformable_79156247265618
// MI455X (gfx1250) — compile-verified
//
#include <hip/hip_runtime.h>
#include <hip/hip_bf16.h>

// CDNA5 / gfx1250: wave32, f32 WMMA 16x16x4. Fused deformable-conv gather-GEMM.
// 32x16 super-tile per wave (2 WMMA tiles share each B fragment). Bilinear
// coefficient tables are software-pipelined one K-step ahead with an
// unroll-by-2 ping-pong (no buffer-rotation copies); the guarded tail step is
// peeled out of the main loop (branch/zero-free staging in steady state).
// All hot-path addressing is 32-bit element offsets off SGPR bases.

typedef __attribute__((ext_vector_type(2))) float v2f;
typedef __attribute__((ext_vector_type(8))) float v8f;

// ---------------------------------------------------------------------------
// Per-layer bilinear coefficient table: for each (k2, p) output tap, the 4
// clamped gather indices and 4 weights (validity folded into the weights,
// matching the reference's v * valid semantics).
// ---------------------------------------------------------------------------
__device__ __forceinline__ void bl_corner(int yi, int xi, float w, int H, int W,
                                          int& id, float& wo) {
    bool v = (yi >= 0) && (yi < H) && (xi >= 0) && (xi < W);
    int yc = yi < 0 ? 0 : (yi > H - 1 ? H - 1 : yi);
    int xc = xi < 0 ? 0 : (xi > W - 1 ? W - 1 : xi);
    id = yc * W + xc;
    wo = v ? w : 0.0f;
}

__global__ void coeff_kernel(const float* __restrict__ off,  // [2*K2, Ho, Wo]
                             int4* __restrict__ idx, float4* __restrict__ wts,
                             int K, int Ho, int Wo, int H, int W) {
    int K2 = K * K, P = Ho * Wo;
    int t = blockIdx.x * blockDim.x + threadIdx.x;
    if (t >= K2 * P) return;
    int k2 = t / P, p = t - k2 * P;
    int oy = p / Wo, ox = p - oy * Wo;
    int ky = k2 / K, kx = k2 - ky * K;
    float dy = off[(2 * k2) * P + p];
    float dx = off[(2 * k2 + 1) * P + p];
    float py = (float)(oy + ky) + dy;
    float px = (float)(ox + kx) + dx;
    float y0f = floorf(py), x0f = floorf(px);
    float wy = py - y0f, wx = px - x0f;
    int y0 = (int)y0f, x0 = (int)x0f;
    int4 id; float4 w4;
    bl_corner(y0,     x0,     (1.f - wy) * (1.f - wx), H, W, id.x, w4.x);
    bl_corner(y0,     x0 + 1, (1.f - wy) * wx,         H, W, id.y, w4.y);
    bl_corner(y0 + 1, x0,     wy * (1.f - wx),         H, W, id.z, w4.z);
    bl_corner(y0 + 1, x0 + 1, wy * wx,                 H, W, id.w, w4.w);
    idx[t] = id;
    wts[t] = w4;
}

// W[Cout][Cin*K2] -> Wt[Kpad][Npad] (row-major, zero padded)
__global__ void prep_weights(const float* __restrict__ w, float* __restrict__ Wmat,
                             int Cout, int CK, int Kpad, int Npad) {
    int t = blockIdx.x * blockDim.x + threadIdx.x;
    if (t >= Kpad * Npad) return;
    int kk = t / Npad, n = t - kk * Npad;
    float v = 0.0f;
    if (kk < CK && n < Cout) v = w[n * CK + kk];
    Wmat[t] = v;
}

// ---------------------------------------------------------------------------
// Fused deformable-conv GEMM, 32x16 super-tile per wave.
// f32 WMMA 16x16x4 fragment layout:
//   A frag: lanes 0-15 hold M=row, K = 2*(lane>>4)+{0,1}
//   B frag: lanes 0-15 hold N=col, K = same split
//   D:      VGPR r -> M = tm + 8*(lane>>4) + r, N = tn + (lane&15)
// ---------------------------------------------------------------------------
struct Stage {
    int4   id0[2], id1[2];
    float4 w0[2],  w1[2];
    float  bf[2];
    int    go0[2], go1[2];
};

__global__ void deform_wmma_gemm(
    const float*  __restrict__ x,    // [B, Cin, HW]
    const int4*   __restrict__ idx,  // [K2*P]
    const float4* __restrict__ wts,  // [K2*P]
    const float*  __restrict__ Wmat, // [Kpad, Npad]
    const float*  __restrict__ bias, // [Cout]
    float*        __restrict__ out,  // [B, Cout, P]
    int P, int Cin, int K2, int CK, int Kpad,
    int Cout, int Npad, int HW, int totalTiles) {
    const int lane = threadIdx.x & 31;
    const int tile = (blockIdx.x * blockDim.x + threadIdx.x) >> 5;
    if (tile >= totalTiles) return;
    const int Ntiles = Npad >> 4;
    const int tmi = tile / Ntiles;
    const int tm  = tmi << 5;                 // 32-row super-tile
    const int tn  = (tile - tmi * Ntiles) << 4;
    const int l15  = lane & 15;
    const int half = lane >> 4;
    const int khalf = half * 2;
    const int col   = tn + l15;               // B-matrix column for this lane

    // Two A rows per lane (two 16-row tiles)
    const int row0 = tm + l15;
    const int row1 = tm + 16 + l15;
    const int b0 = row0 / P, p0 = row0 - b0 * P;
    const int b1 = row1 / P, p1 = row1 - b1 * P;
    const int gb0 = b0 * Cin * HW;            // 32-bit element offsets into x
    const int gb1 = b1 * Cin * HW;

    v8f acc0 = {0.f, 0.f, 0.f, 0.f, 0.f, 0.f, 0.f, 0.f};
    v8f acc1 = {0.f, 0.f, 0.f, 0.f, 0.f, 0.f, 0.f, 0.f};

    // Incremental (kk -> c, k2) tracking; step 4 < K2 (K2 >= 9) per j-slot.
    int kki[2], k2i[2], ci[2];
#pragma unroll
    for (int j = 0; j < 2; ++j) {
        kki[j] = khalf + j;   // < 4 <= K2 always
        k2i[j] = kki[j];
        ci[j]  = 0;
    }

    // Unconditional staging: used for all K-steps where every kk < CK.
    auto stage_full = [&](Stage& S) {
#pragma unroll
        for (int j = 0; j < 2; ++j) {
            const int tb = k2i[j] * P;
            S.id0[j] = idx[tb + p0];
            S.w0[j]  = wts[tb + p0];
            S.id1[j] = idx[tb + p1];
            S.w1[j]  = wts[tb + p1];
            const int coff = ci[j] * HW;
            S.go0[j] = gb0 + coff;
            S.go1[j] = gb1 + coff;
            S.bf[j]  = Wmat[kki[j] * Npad + col];
            kki[j] += 4;
            k2i[j] += 4;
            if (k2i[j] >= K2) { k2i[j] -= K2; ci[j] += 1; }
        }
    };

    // Guarded staging for the single mixed-validity tail step (CK % 4 != 0).
    auto stage_tail = [&](Stage& S) {
#pragma unroll
        for (int j = 0; j < 2; ++j) {
            const int kk = kki[j];
            if (kk < CK) {
                const int tb = k2i[j] * P;
                S.id0[j] = idx[tb + p0];
                S.w0[j]  = wts[tb + p0];
                S.id1[j] = idx[tb + p1];
                S.w1[j]  = wts[tb + p1];
                const int coff = ci[j] * HW;
                S.go0[j] = gb0 + coff;
                S.go1[j] = gb1 + coff;
            } else {
                S.id0[j] = make_int4(0, 0, 0, 0);
                S.id1[j] = make_int4(0, 0, 0, 0);
                S.w0[j]  = make_float4(0.f, 0.f, 0.f, 0.f);
                S.w1[j]  = make_float4(0.f, 0.f, 0.f, 0.f);
                S.go0[j] = gb0;
                S.go1[j] = gb1;
            }
            S.bf[j] = Wmat[kk * Npad + col];  // kk < Kpad
        }
    };

    // Branch-free compute: bilinear gathers + 2 WMMAs sharing the B fragment.
    auto compute = [&](const Stage& S) {
        v2f a0, a1, bf;
#pragma unroll
        for (int j = 0; j < 2; ++j) {
            a0[j] = S.w0[j].x * x[S.go0[j] + S.id0[j].x] +
                    S.w0[j].y * x[S.go0[j] + S.id0[j].y] +
                    S.w0[j].z * x[S.go0[j] + S.id0[j].z] +
                    S.w0[j].w * x[S.go0[j] + S.id0[j].w];
            a1[j] = S.w1[j].x * x[S.go1[j] + S.id1[j].x] +
                    S.w1[j].y * x[S.go1[j] + S.id1[j].y] +
                    S.w1[j].z * x[S.go1[j] + S.id1[j].z] +
                    S.w1[j].w * x[S.go1[j] + S.id1[j].w];
            bf[j] = S.bf[j];
        }
        acc0 = __builtin_amdgcn_wmma_f32_16x16x4_f32(
            false, a0, false, bf, (short)0, acc0, false, false);
        acc1 = __builtin_amdgcn_wmma_f32_16x16x4_f32(
            false, a1, false, bf, (short)0, acc1, false, false);
    };

    const int CKfloor = CK & ~3;
    const int Sfull = CKfloor >> 2;            // full K-steps
    const int Stail = (Kpad - CKfloor) >> 2;   // 0 or 1 tail step

    Stage s0, s1;
    if (Sfull > 0) {
        stage_full(s0);
        if (Sfull > 1) stage_full(s1);
        int s = 0;
        for (; s + 2 <= Sfull; s += 2) {       // ping-pong, no buffer copies
            compute(s0);
            if (s + 2 < Sfull) stage_full(s0);
            compute(s1);
            if (s + 3 < Sfull) stage_full(s1);
        }
        if (s < Sfull) compute(s0);            // odd leftover (unused here)
    }
    if (Stail) {                               // layer-1 only (CK = 9)
        stage_tail(s0);
        compute(s0);
    }

    if (col < Cout) {
        const float bv = bias[col];
        const int mb0 = tm + half * 8;
        const int mb1 = tm + 16 + half * 8;
#pragma unroll
        for (int r = 0; r < 8; ++r) {
            int m  = mb0 + r;
            int bb = m / P;
            int pp = m - bb * P;
            float v = acc0[r] + bv;
            v = v > 0.0f ? v : 0.0f;
            out[(bb * Cout + col) * P + pp] = v;
        }
#pragma unroll
        for (int r = 0; r < 8; ++r) {
            int m  = mb1 + r;
            int bb = m / P;
            int pp = m - bb * P;
            float v = acc1[r] + bv;
            v = v > 0.0f ? v : 0.0f;
            out[(bb * Cout + col) * P + pp] = v;
        }
    }
}

// ---------------------------------------------------------------------------
// Plain f32 WMMA GEMM for the FC head: C = act(A[M,lda] * B[K,ldb] + bias)
// ---------------------------------------------------------------------------
__global__ void fc_wmma_gemm(const float* __restrict__ A, const float* __restrict__ Bm,
                             const float* __restrict__ bias, float* __restrict__ C,
                             int M, int N, int K, int lda, int ldb, int ldc,
                             int relu, int totalTiles) {
    const int lane = threadIdx.x & 31;
    const int tile = (blockIdx.x * blockDim.x + threadIdx.x) >> 5;
    if (tile >= totalTiles) return;
    const int Ntiles = (N + 15) >> 4;
    const int tm = (tile / Ntiles) << 4;
    const int tn = (tile - (tile / Ntiles) * Ntiles) << 4;
    const int l15  = lane & 15;
    const int half = lane >> 4;
    const int row = tm + l15;
    const int col = tn + l15;
    const int khalf = half * 2;

    v8f acc = {0.f, 0.f, 0.f, 0.f, 0.f, 0.f, 0.f, 0.f};
    for (int k = 0; k < K; k += 4) {
        v2f a, bf;
#pragma unroll
        for (int j = 0; j < 2; ++j) {
            int kk = k + khalf + j;
            a[j]  = (kk < K) ? A[row * lda + kk] : 0.0f;
            bf[j] = (kk < K && col < N) ? Bm[kk * ldb + col] : 0.0f;
        }
        acc = __builtin_amdgcn_wmma_f32_16x16x4_f32(
            false, a, false, bf, (short)0, acc, false, false);
    }

    if (col < N) {
        const float bv = bias[col];
        const int mb = tm + half * 8;
#pragma unroll
        for (int r = 0; r < 8; ++r) {
            int m = mb + r;
            float v = acc[r] + bv;
            if (relu) v = v > 0.0f ? v : 0.0f;
            C[m * ldc + col] = v;
        }
    }
}

// act6 [256,4,169] -> flat [256, 676] with spatial perm
__global__ void flatten_perm(const float* __restrict__ act, const int* __restrict__ perm,
                             float* __restrict__ flat) {
    const int F = 4 * 169;
    int t = blockIdx.x * blockDim.x + threadIdx.x;
    if (t >= 256 * F) return;
    int b = t / F, r = t - b * F;
    int c = r / 169, j = r - c * 169;
    flat[t] = act[b * F + c * 169 + perm[j]];
}

// ---------------------------------------------------------------------------
extern "C" void kernel_launch(void* const* d_in, const int* in_sizes, int n_in,
                              void* d_out, int out_size, void* d_ws, size_t ws_size,
                              hipStream_t stream) {
    (void)in_sizes; (void)n_in; (void)out_size; (void)ws_size;

    static const int Cin_[6]  = {1, 16, 32, 16, 16, 8};
    static const int Cout_[6] = {16, 32, 16, 16, 8, 4};
    static const int Kk_[6]   = {3, 3, 5, 7, 5, 3};
    static const int Hin_[6]  = {33, 31, 29, 25, 19, 15};
    static const int Ho_[6]   = {31, 29, 25, 19, 15, 13};
    const int B = 256;

    char* ws = (char*)d_ws;
    // [0, 16MB)   act0  (also reused for flat @0 and h @1MB after the conv stack)
    // [16, 44MB)  act1
    // [44MB+)     coeff idx (512KB), coeff wts (512KB), padded weights (64KB)
    float*  act0 = (float*)(ws);
    float*  act1 = (float*)(ws + (size_t)(16u << 20));
    int4*   cidx = (int4*)(ws + (size_t)(44u << 20));
    float4* cwts = (float4*)(ws + (size_t)(44u << 20) + (512u << 10));
    float*  wmat = (float*)(ws + (size_t)(44u << 20) + (1024u << 10));

    const float* cur = (const float*)d_in[0];  // x: [256,1,33,33]
    for (int i = 0; i < 6; ++i) {
        const int K = Kk_[i], K2 = K * K;
        const int H = Hin_[i], W = H, HW = H * W;
        const int Ho = Ho_[i], P = Ho * Ho;
        const int Cin = Cin_[i], Cout = Cout_[i], CK = Cin * K2;
        const int Kpad = (CK + 3) & ~3;
        const int Npad = (Cout + 15) & ~15;
        const float* off = (const float*)d_in[1 + 3 * i];
        const float* wgt = (const float*)d_in[2 + 3 * i];
        const float* bia = (const float*)d_in[3 + 3 * i];
        float* outp = (i & 1) ? act1 : act0;

        int nco = K2 * P;
        coeff_kernel<<<(nco + 255) / 256, 256, 0, stream>>>(off, cidx, cwts, K, Ho, Ho, H, W);
        int nw = Kpad * Npad;
        prep_weights<<<(nw + 255) / 256, 256, 0, stream>>>(wgt, wmat, Cout, CK, Kpad, Npad);

        // 32-row super-tiles: B*P/32 = 8*P (always integral)
        int totalTiles = (B * P / 32) * (Npad / 16);
        int blocks = (totalTiles + 3) / 4;  // 4 waves (128 threads) per block
        deform_wmma_gemm<<<blocks, 128, 0, stream>>>(
            cur, cidx, cwts, wmat, bia, outp,
            P, Cin, K2, CK, Kpad, Cout, Npad, HW, totalTiles);
        cur = outp;  // [B, Cout, P] feeds next layer
    }

    // FC head: flatten(+perm) -> 676x256 ReLU -> 256x10
    float* flat = act0;                         // act0 free after layer 6
    float* h    = (float*)(ws + (size_t)(1u << 20));
    const int*   perm = (const int*)d_in[23];
    const float* w7 = (const float*)d_in[19];
    const float* b7 = (const float*)d_in[20];
    const float* w8 = (const float*)d_in[21];
    const float* b8 = (const float*)d_in[22];

    int nf = 256 * 676;
    flatten_perm<<<(nf + 255) / 256, 256, 0, stream>>>(cur, perm, flat);

    int t7 = (256 / 16) * (256 / 16);  // M=256, N=256
    fc_wmma_gemm<<<(t7 + 3) / 4, 128, 0, stream>>>(
        flat, w7, b7, h, 256, 256, 676, 676, 256, 256, 1, t7);

    int t8 = (256 / 16) * 1;           // M=256, N=10 (one padded N tile)
    fc_wmma_gemm<<<(t8 + 3) / 4, 128, 0, stream>>>(
        h, w8, b8, (float*)d_out, 256, 10, 256, 256, 10, 10, 0, t8);
}